// LIFNeuron_72430328479801
// MI455X (gfx1250) — compile-verified
//
#include <hip/hip_runtime.h>
#include <stdint.h>

// ---------------------------------------------------------------------------
// LIF neuron forward, T=50 steps, elementwise over N = 16*32*64*64 = 2,097,152
// d_out layout (float32, concatenated in return order):
//   [0        , 50N)  memHist  (pre-reset membrane, per step)
//   [50N      , 100N) spkHist  (0/1 spikes, per step)
//   [100N     , 101N) final mem (post-reset)
//   [101N     , 102N) encoded = sum(spk)/50
// Store-bandwidth bound: ~864 MB traffic -> ~37us floor at 23.3 TB/s.
// ---------------------------------------------------------------------------

typedef float v4f __attribute__((ext_vector_type(4)));

#define T_STEPS   50
#define U_REST    (-75.0f)
#define THRESH    (-55.0f)
#define REF_TIME  2.0f
#define INV_TAU   0.1f      // 1.0 / TAU_MEM(10.0), rounded to f32 exactly as in ref
#define RES_MEM   0.1f

__global__ __launch_bounds__(256) void lif_forward_kernel(
    const float* __restrict__ x,
    float* __restrict__ out,
    int n)
{
    // 256 threads * 4 elements = 1024 elements per block
    __shared__ float sx[256 * 4] __attribute__((aligned(16)));

    const int tid = threadIdx.x;
    const size_t i4 = (size_t)blockIdx.x * 1024u + (size_t)tid * 4u;
    if (i4 >= (size_t)n) return;   // n is a multiple of 1024 in practice

    // ---- CDNA5 async data mover: global -> LDS (128b per lane), then wait ---
    // LDS address is taken from the first VGPR operand; low 32 bits of a generic shared pointer
    // are the LDS byte offset.
    {
        const float* gsrc = x + i4;
        unsigned lds_off = (unsigned)(uintptr_t)(&sx[tid * 4]);
        asm volatile("global_load_async_to_lds_b128 %0, %1, off"
                     :: "v"(lds_off), "v"(gsrc)
                     : "memory");
        asm volatile("s_wait_asynccnt 0" ::: "memory");
    }
    const v4f xv = *(const v4f*)(&sx[tid * 4]);

    const size_t N = (size_t)n;
    float* __restrict__ memH = out;                 // [50*N]
    float* __restrict__ spkH = out + 50u * N;       // [50*N]
    float* __restrict__ memF = out + 100u * N;      // [N]
    float* __restrict__ enc  = out + 101u * N;      // [N]

    float xl[4]  = { xv.x, xv.y, xv.z, xv.w };
    float m[4]   = { U_REST, U_REST, U_REST, U_REST };
    float rf[4]  = { 0.f, 0.f, 0.f, 0.f };
    float acc[4] = { 0.f, 0.f, 0.f, 0.f };

    float* mrow = memH + i4;
    float* srow = spkH + i4;

    for (int t = 0; t < T_STEPS; ++t) {
        v4f mh, sp;
        float mhv[4], spv[4];
        #pragma unroll
        for (int c = 0; c < 4; ++c) {
            // ref_mask blocks input while refractory; inv decrements counter
            const float rm  = (rf[c] > 0.0f) ? 0.0f : 1.0f;
            const float rmi = 1.0f - rm;
            // leaky integration: dv = (-(mem - uRest) + x*resMem*refMask)/tau
            m[c] = m[c] + INV_TAU * (-(m[c] - U_REST) + xl[c] * RES_MEM * rm);
            const float spk = (m[c] >= THRESH) ? 1.0f : 0.0f;
            rf[c] = (spk == 1.0f) ? REF_TIME : rf[c];
            mhv[c] = m[c];                 // history recorded PRE-reset
            spv[c] = spk;
            m[c]  = m[c] - spk * THRESH;   // 'subtract' reset mode
            rf[c] = rf[c] - rmi;           // decrement only previously-refractory
            acc[c] += spk;
        }
        mh.x = mhv[0]; mh.y = mhv[1]; mh.z = mhv[2]; mh.w = mhv[3];
        sp.x = spv[0]; sp.y = spv[1]; sp.z = spv[2]; sp.w = spv[3];

        // Streaming 856MB output >> 192MB L2: non-temporal 128-bit stores.
        __builtin_nontemporal_store(mh, (v4f*)mrow);
        __builtin_nontemporal_store(sp, (v4f*)srow);
        mrow += N;
        srow += N;
    }

    v4f mf, ev;
    mf.x = m[0]; mf.y = m[1]; mf.z = m[2]; mf.w = m[3];
    // true divide by 50.0f to match reference rounding (50 has no exact recip)
    ev.x = acc[0] / 50.0f; ev.y = acc[1] / 50.0f;
    ev.z = acc[2] / 50.0f; ev.w = acc[3] / 50.0f;
    __builtin_nontemporal_store(mf, (v4f*)(memF + i4));
    __builtin_nontemporal_store(ev, (v4f*)(enc  + i4));
}

extern "C" void kernel_launch(void* const* d_in, const int* in_sizes, int n_in,
                              void* d_out, int out_size, void* d_ws, size_t ws_size,
                              hipStream_t stream) {
    const float* x = (const float*)d_in[0];
    float* out = (float*)d_out;
    const int n = in_sizes[0];                 // 2,097,152
    const int blocks = (n + 1023) / 1024;      // 1024 elements per 256-thread block
    lif_forward_kernel<<<blocks, 256, 0, stream>>>(x, out, n);
}